// Attention_56384330662199
// MI455X (gfx1250) — compile-verified
//
#include <hip/hip_runtime.h>
#include <hip/hip_bf16.h>
#include <cstdint>
#include <cstddef>

// MI455X / gfx1250, wave32.
// B=32, S=4096, D=512 attention-pooling. Pure HBM-bandwidth bound (x = 256MB).
// Single sweep of x (scores via V_WMMA_F32_16X16X4_F32) + L2-hot re-read for
// the weighted sum; tiny combine kernel. W staged to LDS via async-to-LDS.

typedef __attribute__((ext_vector_type(2))) float v2f;
typedef __attribute__((ext_vector_type(8))) float v8f;
typedef __attribute__((ext_vector_type(2))) int   v2i;

#define B_     32
#define S_     4096
#define D_     512
#define CHUNKS 32          // sequence chunks per batch
#define CH     128         // rows per workgroup (S_/CHUNKS)
#define WAVES  8
#define TPB    256

__device__ __forceinline__ float fast_tanh(float v) {
#if __has_builtin(__builtin_amdgcn_tanhf)
    return __builtin_amdgcn_tanhf(v);      // v_tanh_f32 (TRANS op)
#else
    return tanhf(v);
#endif
}

// ---------------------------------------------------------------------------
// Pass 1: one workgroup per (batch, 128-row chunk).
//  Phase A: 8 waves x 16 rows each; V_WMMA_F32_16X16X4_F32 computes 16 row
//           dots per wave (B operand = W chunk replicated across columns, so
//           every D column equals the row-dot -> no cross-lane reduction).
//  Phase B: re-read the (L2-hot) 256KB tile coalesced; acc[c] += e[s]*x[s,c].
// ---------------------------------------------------------------------------
__global__ __launch_bounds__(TPB) void attn_pass1(
    const float* __restrict__ x,     // (B,S,D)
    const float* __restrict__ W,     // (D,1)
    const float* __restrict__ bias,  // (S,1)
    float* __restrict__ e_scr,       // (B,S)  exp(score)
    float* __restrict__ pAcc,        // (B*CHUNKS, D) partial weighted sums
    float* __restrict__ pL)          // (B*CHUNKS)    partial exp-sums
{
    __shared__ float ldsW[D_];
    __shared__ float ldsE[CH];
    __shared__ float ldsL[WAVES];

    const int tid   = threadIdx.x;
    const int lane  = tid & 31;
    const int w     = tid >> 5;       // wave id 0..7
    const int chunk = blockIdx.x;     // 0..31
    const int b     = blockIdx.y;     // 0..31
    const int r0    = chunk * CH;

    // ---- Stage W (512 floats) into LDS ----------------------------------
#if __has_builtin(__builtin_amdgcn_global_load_async_to_lds_b64)
    {
        // GLOBAL_LOAD_ASYNC_TO_LDS_B64: 8B per lane, coalesced, ASYNCcnt.
        // Builtin signature (from hipcc diagnostic): param0 = AS(1) v2i*.
        // Generic->AS1 keeps the 64-bit VA; generic->AS3 is the low 32 bits
        // (flat-LDS mapping truncates to addr[31:0]); route through uintptr_t.
        __attribute__((address_space(1))) v2i* gsrc =
            (__attribute__((address_space(1))) v2i*)(uintptr_t)(W + 2 * tid);
        __attribute__((address_space(3))) v2i* ldst =
            (__attribute__((address_space(3))) v2i*)(uint32_t)(uintptr_t)(&ldsW[2 * tid]);
        __builtin_amdgcn_global_load_async_to_lds_b64(gsrc, ldst, 0, 0);
#if __has_builtin(__builtin_amdgcn_s_wait_asynccnt)
        __builtin_amdgcn_s_wait_asynccnt(0);
#else
        asm volatile("s_wait_asynccnt 0x0" ::: "memory");
#endif
    }
#else
    {
        v2f t = *(const v2f*)(W + 2 * tid);
        *(v2f*)(&ldsW[2 * tid]) = t;
    }
#endif
    __syncthreads();

    // ---- Phase A: scores for rows [r0 + 16w, r0 + 16w + 16) --------------
    const int half = lane >> 4;        // 0: rows 0..7 of tile, 1: rows 8..15
    const int koff = half * 2;         // K sub-offset per A/B f32 layout
    const int mrow = lane & 15;
    const float* xrow =
        x + ((size_t)b * S_ + (size_t)(r0 + 16 * w + mrow)) * D_;

    v8f c = {};   // D matrix: every column identical = per-row dot
#if __has_builtin(__builtin_amdgcn_wmma_f32_16x16x4_f32)
    for (int t = 0; t < D_ / 4; ++t) {
        const int k = 4 * t + koff;
        v2f a  = *(const v2f*)(xrow + k);       // A[m=mrow][k..k+1]
        v2f bb = *(const v2f*)(&ldsW[k]);       // B[k..k+1][n] = W (all n)
        c = __builtin_amdgcn_wmma_f32_16x16x4_f32(
                false, a, false, bb, (short)0, c, false, false);
    }
#else
    // Compile-safe fallback with identical D-matrix semantics.
    for (int i = 0; i < 8; ++i) {
        const float* row =
            x + ((size_t)b * S_ + (size_t)(r0 + 16 * w + half * 8 + i)) * D_;
        float d = 0.f;
        for (int k = 0; k < D_; ++k) d += row[k] * ldsW[k];
        c[i] = d;
    }
#endif

    // Each lane now holds 8 row-dots: rows rbase..rbase+7.
    const int rbase = r0 + 16 * w + half * 8;
    float evals[8];
    float lsum = 0.f;
#pragma unroll
    for (int i = 0; i < 8; ++i) {
        const float sc = fast_tanh(c[i] + bias[rbase + i]);  // |sc| < 1
        const float e  = __expf(sc);                         // bounded, safe
        evals[i] = e;
        lsum += e;
    }
    // lanes 0-15 carry rows 0..7, lanes 16-31 rows 8..15 -> one xor-shuffle.
    const float l_wave = lsum + __shfl_xor(lsum, 16);

    if (mrow == 0) {   // lane 0 and lane 16 are the representatives
#pragma unroll
        for (int i = 0; i < 8; ++i) {
            ldsE[16 * w + half * 8 + i]       = evals[i];
            e_scr[(size_t)b * S_ + rbase + i] = evals[i];
        }
        if (half == 0) ldsL[w] = l_wave;
    }
    __syncthreads();

    // ---- Phase B: weighted accumulation over the (L2-hot) tile -----------
    v2f acc = {0.f, 0.f};
    const float* xc = x + ((size_t)b * S_ + r0) * D_ + 2 * tid;
#pragma unroll 4
    for (int s = 0; s < CH; ++s) {
        const float e = ldsE[s];
        v2f xv = *(const v2f*)(xc + (size_t)s * D_);
        acc.x += e * xv.x;
        acc.y += e * xv.y;
    }
    const size_t wg = (size_t)b * CHUNKS + chunk;
    *(v2f*)(&pAcc[wg * D_ + 2 * tid]) = acc;
    if (tid == 0) {
        float L = 0.f;
        for (int i = 0; i < WAVES; ++i) L += ldsL[i];
        pL[wg] = L;
    }
}

// ---------------------------------------------------------------------------
// Pass 2: one workgroup per batch. Reduce partials, emit context + weights.
// d_out layout: context (B*D floats) then weights (B*S floats).
// ---------------------------------------------------------------------------
__global__ __launch_bounds__(TPB) void attn_pass2(
    const float* __restrict__ e_scr,
    const float* __restrict__ pAcc,
    const float* __restrict__ pL,
    float* __restrict__ out)
{
    __shared__ float tmp[CHUNKS];
    __shared__ float invl_s;
    const int tid = threadIdx.x;
    const int b   = blockIdx.x;

    if (tid < CHUNKS) tmp[tid] = pL[(size_t)b * CHUNKS + tid];
    __syncthreads();
    if (tid == 0) {
        float t = 0.f;
        for (int i = 0; i < CHUNKS; ++i) t += tmp[i];
        invl_s = 1.0f / t;
    }
    __syncthreads();
    const float inv_l = invl_s;

    // context[b, 2*tid .. 2*tid+1]
    v2f ctx = {0.f, 0.f};
    for (int ch = 0; ch < CHUNKS; ++ch) {
        v2f p = *(const v2f*)(&pAcc[((size_t)b * CHUNKS + ch) * D_ + 2 * tid]);
        ctx.x += p.x;
        ctx.y += p.y;
    }
    ctx.x *= inv_l;
    ctx.y *= inv_l;
    *(v2f*)(&out[(size_t)b * D_ + 2 * tid]) = ctx;

    // weights[b, s] = e / l
    float* wout       = out + (size_t)B_ * D_ + (size_t)b * S_;
    const float* esrc = e_scr + (size_t)b * S_;
#pragma unroll
    for (int k = 0; k < S_ / TPB; ++k) {
        const int s = tid + k * TPB;
        wout[s] = esrc[s] * inv_l;
    }
}

// ---------------------------------------------------------------------------
extern "C" void kernel_launch(void* const* d_in, const int* in_sizes, int n_in,
                              void* d_out, int out_size, void* d_ws, size_t ws_size,
                              hipStream_t stream)
{
    const float* x    = (const float*)d_in[0];   // (B,S,D) f32
    const float* W    = (const float*)d_in[1];   // (D,1)   f32
    const float* bias = (const float*)d_in[2];   // (S,1)   f32
    float* out = (float*)d_out;

    char* ws = (char*)d_ws;
    float* e_scr = (float*)ws;                                   // B*S floats
    float* pAcc  = (float*)(ws + (size_t)B_ * S_ * sizeof(float));
    float* pL    = (float*)(ws + (size_t)B_ * S_ * sizeof(float)
                               + (size_t)B_ * CHUNKS * D_ * sizeof(float));

    dim3 g1(CHUNKS, B_);
    attn_pass1<<<g1, TPB, 0, stream>>>(x, W, bias, e_scr, pAcc, pL);
    attn_pass2<<<B_, TPB, 0, stream>>>(e_scr, pAcc, pL, out);
}